// BaseEmbedding_51582557225399
// MI455X (gfx1250) — compile-verified
//
#include <hip/hip_runtime.h>

typedef __attribute__((ext_vector_type(2))) float v2f;
typedef __attribute__((ext_vector_type(8))) float v8f;

// Mean-pooling over 32 ragged segments ([32,64,96,64] x 8, 2048 fields) of
// emb [B, 2048, 64] f32 -> out [B, 32, 64] f32, expressed as a per-batch GEMM
// W(32x2048) x E_b(2048x64) on V_WMMA_F32_16X16X4_F32.
//
// Tile map: one block per batch, 8 waves = {mtile 0..1} x {ntile 0..3}.
// mtile m owns segments [16m,16m+16) == fields [1024m, 1024m+1024).
// Each K-chunk of 4 fields lies in one segment, so the A operand for that
// chunk is uniform: av = (lane%16 == s_local) ? 1/size : 0 in both K slots.
// Since all K-weights in a chunk are equal, the result is invariant to the
// K-permutation of the B operand layout; only the N(lane) mapping matters.

__global__ __launch_bounds__(256)
void mean_pool_wmma(const float* __restrict__ emb, float* __restrict__ out,
                    int batch) {
  const int b     = blockIdx.x;
  const int wave  = threadIdx.x >> 5;   // 0..7
  const int lane  = threadIdx.x & 31;
  const int mtile = wave >> 2;          // 0..1  (segment halves)
  const int ntile = wave & 3;           // 0..3  (embed-dim 16-col tiles)
  const int half  = lane >> 4;          // 0..1  (K sub-rows of the chunk)
  const int n     = lane & 15;          // N = embed column within tile
  if (b >= batch) return;

  // B operand base: row (f_rel + 2*half), column n of E_b restricted to
  // fields [1024*mtile, ...), cols [16*ntile, ...).  Row stride = 64 floats.
  const float* p = emb
      + ((size_t)b * 2048 + (size_t)mtile * 1024 + (size_t)half * 2) * 64
      + (size_t)ntile * 16 + n;

  v8f acc = {};  // 16x16 f32 C/D accumulator (8 VGPRs)

  const int sizes4[4] = {32, 64, 96, 64};   // pattern repeats 4x per half
  int s_local = 0;

  for (int rep = 0; rep < 4; ++rep) {
    #pragma unroll
    for (int j = 0; j < 4; ++j) {
      const int   sz = sizes4[j];
      const float w  = 1.0f / (float)sz;
      // One nonzero A row per chunk; same weight in both K slots of this lane.
      const float av = (n == s_local) ? w : 0.0f;
      v2f a; a[0] = av; a[1] = av;

      const int steps = sz >> 2;            // 4 fields per WMMA K-chunk
      #pragma unroll 8
      for (int t = 0; t < steps; ++t) {
        v2f bv;
        bv[0] = p[0];    // row f_rel + 2*half,     col n
        bv[1] = p[64];   // row f_rel + 2*half + 1, col n
        acc = __builtin_amdgcn_wmma_f32_16x16x4_f32(
            /*neg_a=*/false, a, /*neg_b=*/false, bv,
            /*c_mod=*/(short)0, acc, /*reuse_a=*/false, /*reuse_b=*/false);
        p += 4 * 64;     // advance 4 fields
      }
      ++s_local;
    }
  }

  // D layout: VGPR r -> M = r + 8*half, N = n.  s = 16*mtile + M, d = 16*ntile + N.
  float* obase = out
      + ((size_t)b * 32 + (size_t)mtile * 16 + (size_t)half * 8) * 64
      + (size_t)ntile * 16 + n;
  #pragma unroll
  for (int r = 0; r < 8; ++r) {
    obase[(size_t)r * 64] = acc[r];
  }
}

extern "C" void kernel_launch(void* const* d_in, const int* in_sizes, int n_in,
                              void* d_out, int out_size, void* d_ws, size_t ws_size,
                              hipStream_t stream) {
  const float* emb = (const float*)d_in[0];
  float* out = (float*)d_out;
  const int batch = in_sizes[0] / (2048 * 64);   // 1024 for the reference setup
  dim3 grid((unsigned)batch);
  dim3 block(256);
  hipLaunchKernelGGL(mean_pool_wmma, grid, block, 0, stream, emb, out, batch);
}